// Q_81097572483268
// MI455X (gfx1250) — compile-verified
//
#include <hip/hip_runtime.h>

#define NN  1024
#define DIM 64

typedef __attribute__((ext_vector_type(2))) float v2f;
typedef __attribute__((ext_vector_type(8))) float v8f;

// ---------------------------------------------------------------------------
// Kernel 1: Pos[i] = sum_k relu(g[i,k]);  Neg[i] = sum_k min(g[i,k],0)
// Pure HBM-bandwidth pass over the 4MB graph (roofline-limiting term).
// ---------------------------------------------------------------------------
__global__ void rowsum_kernel(const float* __restrict__ g,
                              float* __restrict__ Pos,
                              float* __restrict__ Neg) {
  __shared__ float sp[8], sn[8];
  const int row = blockIdx.x;
  const float4* gr = (const float4*)(g + (size_t)row * NN);
  float4 v = gr[threadIdx.x];                       // 256 threads * 16B = one row
  float p = fmaxf(v.x, 0.f) + fmaxf(v.y, 0.f) + fmaxf(v.z, 0.f) + fmaxf(v.w, 0.f);
  float n = fminf(v.x, 0.f) + fminf(v.y, 0.f) + fminf(v.z, 0.f) + fminf(v.w, 0.f);
  // wave32 tree reduction
  for (int off = 16; off > 0; off >>= 1) {
    p += __shfl_down(p, off);
    n += __shfl_down(n, off);
  }
  const int lane = threadIdx.x & 31, w = threadIdx.x >> 5;
  if (lane == 0) { sp[w] = p; sn[w] = n; }
  __syncthreads();
  if (threadIdx.x == 0) {
    float P = 0.f, Ng = 0.f;
    for (int i = 0; i < 8; ++i) { P += sp[i]; Ng += sn[i]; }
    Pos[row] = P;
    Neg[row] = Ng;
  }
}

// ---------------------------------------------------------------------------
// Kernel 2: fused structure2vec rounds + Q(v).  One 32-wave workgroup; u lives
// in LDS (256KB of the WGP's 320KB).  GEMM rounds use V_WMMA_F32_16X16X4_F32.
// ---------------------------------------------------------------------------
#define SMEM_FLOATS (NN*DIM + DIM*DIM + 3*NN + 3*DIM + DIM + 16*DIM + DIM)
#define SMEM_BYTES  (SMEM_FLOATS * 4)

__global__ __launch_bounds__(1024) void s2v_kernel(
    const float* __restrict__ x,  const float* __restrict__ W1,
    const float* __restrict__ W2, const float* __restrict__ W3,
    const float* __restrict__ W4, const float* __restrict__ W5,
    const float* __restrict__ W6, const float* __restrict__ W7,
    const int*  __restrict__ vp,
    const float* __restrict__ Pos, const float* __restrict__ Neg,
    float* __restrict__ out) {
  extern __shared__ float smem[];
  float* sU    = smem;                 // [NN][DIM]  node embeddings u
  float* sW2   = sU + NN * DIM;        // [DIM][DIM] row-major theta2
  float* sX    = sW2 + DIM * DIM;      // [NN]
  float* sP    = sX + NN;              // [NN]  Pos
  float* sNg   = sP + NN;              // [NN]  Neg
  float* sW1   = sNg + NN;             // [DIM]
  float* sAp   = sW1 + DIM;            // [DIM]  W3 @ relu(W4)
  float* sAn   = sAp + DIM;            // [DIM]  W3 @ min(W4,0)
  float* sSum  = sAn + DIM;            // [DIM]  column sums of u
  float* sPart = sSum + DIM;           // [16][DIM] reduction staging
  float* sRed  = sPart + 16 * DIM;     // [DIM]  final dot staging

  const int tid = threadIdx.x;

  // ---- setup: stage invariants into LDS; collapse t3 to rank-2 ----
  for (int i = tid; i < DIM * DIM; i += 1024) sW2[i] = W2[i];
  sX[tid] = x[tid]; sP[tid] = Pos[tid]; sNg[tid] = Neg[tid];
  if (tid < DIM) {
    sW1[tid] = W1[tid];
    float ap = 0.f, an = 0.f;
    for (int d = 0; d < DIM; ++d) {
      float w3 = W3[tid * DIM + d];
      ap += w3 * fmaxf(W4[d], 0.f);
      an += w3 * fminf(W4[d], 0.f);
    }
    sAp[tid] = ap; sAn[tid] = an;
  }
  __syncthreads();

  // ---- round 1 (u==0 => t2==0): u = relu(t1 + t3) ----
  {
    const float xi = sX[tid], pi = sP[tid], ni = sNg[tid];
    for (int d = 0; d < DIM; ++d)
      sU[tid * DIM + d] = fmaxf(xi * sW1[d] + pi * sAp[d] + ni * sAn[d], 0.f);
  }
  __syncthreads();

  const int wave = tid >> 5;
  const int lane = tid & 31;
  const int ml   = lane & 15;            // M (or N) index within 16
  const int kb   = (lane < 16) ? 0 : 2;  // K sub-offset for A/B frags
  const int mo   = (lane < 16) ? 0 : 8;  // M offset for C/D frags

  // ---- rounds 2..4: u = relu(t13 + (sum(u) - u) @ W2^T) ----
  for (int round = 1; round < 4; ++round) {
    // column sums of u -> sSum  (two-stage: 16 partials x 64 cols)
    {
      const int d = tid & 63, r0 = tid >> 6;
      float s = 0.f;
      for (int j = 0; j < 64; ++j) s += sU[(r0 + j * 16) * DIM + d];
      sPart[r0 * DIM + d] = s;
    }
    __syncthreads();
    if (tid < DIM) {
      float s = 0.f;
      for (int j = 0; j < 16; ++j) s += sPart[j * DIM + tid];
      sSum[tid] = s;
    }
    __syncthreads();

    // GEMM via f32 WMMA.  Each wave owns 16-row tiles {wave, wave+32};
    // a tile's A-frags read only that tile's rows of sU, so in-place
    // relu-fused writeback is race-free without double buffering.
    for (int tile = wave; tile < 64; tile += 32) {
      const int row0 = tile * 16;
      const float* uRow = sU + (row0 + ml) * DIM;
      v8f acc[4] = {v8f{}, v8f{}, v8f{}, v8f{}};
#pragma unroll
      for (int k0 = 0; k0 < DIM; k0 += 4) {
        const v2f uv = *(const v2f*)(uRow + k0 + kb);
        const v2f sv = *(const v2f*)(sSum + k0 + kb);
        v2f a; a.x = sv.x - uv.x; a.y = sv.y - uv.y;
#pragma unroll
        for (int nt = 0; nt < 4; ++nt) {
          const v2f b = *(const v2f*)(sW2 + (nt * 16 + ml) * DIM + k0 + kb);
          acc[nt] = __builtin_amdgcn_wmma_f32_16x16x4_f32(
              false, a, false, b, (short)0, acc[nt], false, false);
        }
      }
      // epilogue: u = relu(acc + x*W1 + Pos*ap + Neg*an), written in place
#pragma unroll
      for (int r = 0; r < 8; ++r) {
        const int m = row0 + r + mo;
        const float xm = sX[m], pm = sP[m], nm = sNg[m];
#pragma unroll
        for (int nt = 0; nt < 4; ++nt) {
          const int n = nt * 16 + ml;
          sU[m * DIM + n] =
              fmaxf(acc[nt][r] + xm * sW1[n] + pm * sAp[n] + nm * sAn[n], 0.f);
        }
      }
    }
    __syncthreads();
  }

  // ---- final: Q = W5 . relu([W6 @ sum(u) ; W7 @ u[v]]) ----
  {
    const int d = tid & 63, r0 = tid >> 6;
    float s = 0.f;
    for (int j = 0; j < 64; ++j) s += sU[(r0 + j * 16) * DIM + d];
    sPart[r0 * DIM + d] = s;
  }
  __syncthreads();
  if (tid < DIM) {
    float s = 0.f;
    for (int j = 0; j < 16; ++j) s += sPart[j * DIM + tid];
    sSum[tid] = s;
  }
  __syncthreads();
  if (tid < DIM) {
    const int v = vp[0];
    float h1 = 0.f, h2 = 0.f;
    for (int d = 0; d < DIM; ++d) {
      h1 += W6[tid * DIM + d] * sSum[d];
      h2 += W7[tid * DIM + d] * sU[v * DIM + d];
    }
    h1 = fmaxf(h1, 0.f);
    h2 = fmaxf(h2, 0.f);
    sRed[tid] = W5[tid] * h1 + W5[DIM + tid] * h2;
  }
  __syncthreads();
  if (tid == 0) {
    float q = 0.f;
    for (int j = 0; j < DIM; ++j) q += sRed[j];
    out[0] = q;
  }
}

// ---------------------------------------------------------------------------
extern "C" void kernel_launch(void* const* d_in, const int* in_sizes, int n_in,
                              void* d_out, int out_size, void* d_ws, size_t ws_size,
                              hipStream_t stream) {
  const float* x     = (const float*)d_in[0];
  const float* graph = (const float*)d_in[1];
  const float* W1    = (const float*)d_in[2];
  const float* W2    = (const float*)d_in[3];
  const float* W3    = (const float*)d_in[4];
  const float* W4    = (const float*)d_in[5];
  const float* W5    = (const float*)d_in[6];
  const float* W6    = (const float*)d_in[7];
  const float* W7    = (const float*)d_in[8];
  const int*   v     = (const int*)d_in[9];
  float* out = (float*)d_out;

  float* Pos = (float*)d_ws;
  float* Neg = Pos + NN;

  // allow >64KB dynamic LDS (WGP has 320KB)
  (void)hipFuncSetAttribute((const void*)s2v_kernel,
                            hipFuncAttributeMaxDynamicSharedMemorySize,
                            (int)SMEM_BYTES);

  rowsum_kernel<<<NN, 256, 0, stream>>>(graph, Pos, Neg);
  s2v_kernel<<<1, 1024, SMEM_BYTES, stream>>>(x, W1, W2, W3, W4, W5, W6, W7,
                                              v, Pos, Neg, out);
}